// AsymQuantizer_38594576122187
// MI455X (gfx1250) — compile-verified
//
#include <hip/hip_runtime.h>
#include <hip/hip_fp16.h>

// Per-row asymmetric int4 quantization (QuaRot asym_quant semantics).
//   x: [R, C] fp32, C = 4096, R = 16384
//   out = [ q (R*C fp32) | scale (R fp32) | zeros (R, fp16-rounded, stored fp32) ]
//
// Memory-bound: ~537 MB traffic -> ~23 us floor at 23.3 TB/s.
// CDNA5 path: async global->LDS DMA staging (ASYNCcnt), wave32 shfl_xor
// reductions, nontemporal b128 streaming stores.

typedef float v4f __attribute__((ext_vector_type(4)));
typedef int   v4i __attribute__((ext_vector_type(4)));

// Pointer types matching the gfx1250 async-LDS builtin signature:
//   (int4 addrspace(1)*, int4 addrspace(3)*, imm offset, imm cpol)
typedef __attribute__((address_space(1))) v4i* gptr_v4i;
typedef __attribute__((address_space(3))) v4i* lptr_v4i;

#define TPB    256            // 8 wave32 per block
#define COLS   4096           // row length (compile-time, matches reference)
#define CHUNKS (COLS / 4)     // 1024 float4 chunks per row
#define CPT    (CHUNKS / TPB) // 4 chunks (16 floats) per thread
#define MAXQF  15.0f

#if defined(__HIP_DEVICE_COMPILE__) && defined(__gfx1250__) && \
    __has_builtin(__builtin_amdgcn_global_load_async_to_lds_b128)
#define HAVE_ASYNC_LDS 1
#else
#define HAVE_ASYNC_LDS 0
#endif

__global__ __launch_bounds__(TPB) void asym_quant_i4_rowwise(
    const float* __restrict__ x,
    float* __restrict__ q,
    float* __restrict__ scale_out,
    float* __restrict__ zero_out)
{
    __shared__ v4f   srow[CHUNKS];        // 16 KB staging for one row
    __shared__ float wmin[TPB / 32];
    __shared__ float wmax[TPB / 32];

    const int    t       = threadIdx.x;
    const size_t rowbase = (size_t)blockIdx.x * COLS;
    const v4f* __restrict__ xr = (const v4f*)(x + rowbase);

    // ---- Stage the row into LDS ------------------------------------------
#if HAVE_ASYNC_LDS
    // Async DMA: global -> LDS without flowing through VGPRs.
    // Each thread issues CPT b128 transfers, tracked by ASYNCcnt.
#pragma unroll
    for (int c = 0; c < CPT; ++c) {
        const int ch = c * TPB + t;
        __builtin_amdgcn_global_load_async_to_lds_b128(
            (gptr_v4i)(xr + ch),       // global source (AS1 int4*)
            (lptr_v4i)(srow + ch),     // LDS destination (AS3 int4*)
            /*offset=*/0, /*cpol=*/0);
    }
#if __has_builtin(__builtin_amdgcn_s_wait_asynccnt)
    __builtin_amdgcn_s_wait_asynccnt(0);
#else
    asm volatile("s_wait_asynccnt 0" ::: "memory");
#endif
#else
    // Fallback: plain coalesced b128 copy through VGPRs.
#pragma unroll
    for (int c = 0; c < CPT; ++c) {
        const int ch = c * TPB + t;
        srow[ch] = xr[ch];
    }
#endif
    __syncthreads();   // all 8 waves' staging visible

    // ---- Per-thread min/max; values stay in registers for the quant pass --
    v4f   v[CPT];
    float mn =  3.402823466e+38f;
    float mx = -3.402823466e+38f;
#pragma unroll
    for (int c = 0; c < CPT; ++c) {
        v[c] = srow[c * TPB + t];          // one ds_load_b128 each
#pragma unroll
        for (int k = 0; k < 4; ++k) {
            mn = fminf(mn, v[c][k]);
            mx = fmaxf(mx, v[c][k]);
        }
    }

    // ---- wave32 butterfly reduction (CDNA5 is wave32-only) ----------------
#pragma unroll
    for (int off = 16; off > 0; off >>= 1) {
        mn = fminf(mn, __shfl_xor(mn, off, 32));
        mx = fmaxf(mx, __shfl_xor(mx, off, 32));
    }
    if ((t & 31) == 0) { wmin[t >> 5] = mn; wmax[t >> 5] = mx; }
    __syncthreads();
#pragma unroll
    for (int w = 0; w < TPB / 32; ++w) {
        mn = fminf(mn, wmin[w]);
        mx = fmaxf(mx, wmax[w]);
    }

    // ---- Row constants ----------------------------------------------------
    const float scale = (mx - mn) / MAXQF;               // exact, once per row
    const float zraw  = rintf(-mn / scale);              // round-half-even == jnp.round
    const float zf    = __half2float(__float2half_rn(zraw)); // fp16 zero-point (exact: int 0..15)
    const float rs    = 1.0f / scale;                    // 1 exact divide/row; mul per element

    // ---- Quantize from registers; nontemporal streaming stores ------------
    float* __restrict__ qrow = q + rowbase;
#pragma unroll
    for (int c = 0; c < CPT; ++c) {
        const int ch = c * TPB + t;
        v4f o;
#pragma unroll
        for (int k = 0; k < 4; ++k) {
            const float qv = rintf(v[c][k] * rs) + zf;
            o[k] = fminf(fmaxf(qv, 0.0f), MAXQF);
        }
        __builtin_nontemporal_store(o, (v4f*)qrow + ch); // write-once data: bypass temporal caching
    }

    if (t == 0) {
        scale_out[blockIdx.x] = scale;
        zero_out[blockIdx.x]  = zf;
    }
}

extern "C" void kernel_launch(void* const* d_in, const int* in_sizes, int n_in,
                              void* d_out, int out_size, void* d_ws, size_t ws_size,
                              hipStream_t stream)
{
    (void)n_in; (void)d_ws; (void)ws_size; (void)out_size;

    const float* x    = (const float*)d_in[0];
    const int    n    = in_sizes[0];       // R * 4096
    const int    rows = n / COLS;          // 16384

    float* qout  = (float*)d_out;                         // [R*C]
    float* scale = qout + (size_t)rows * COLS;            // [R]
    float* zeros = scale + rows;                          // [R]

    asym_quant_i4_rowwise<<<rows, TPB, 0, stream>>>(x, qout, scale, zeros);
}